// FeaturesEmbedding_26422638805035
// MI455X (gfx1250) — compile-verified
//
#include <hip/hip_runtime.h>

// Sparse (COO, sorted rows) x dense embedding:  out[r,:] += vals[i] * E[cols[i],:]
// BATCH=16384, EMBED_DIM=64, f32. Memory-bound (AI ~0.45 F/B; ~230MB -> ~10us
// floor at 23.3 TB/s). WMMA f32 16x16x4 acts as a zero-atomic segment-sum engine:
// A = row-indicator*vals (16x4), B = gathered E dims (4x16), D accumulates a
// 16x16 output tile per wave. Row tiles own disjoint nnz spans (rows sorted +
// binary search) so no atomics and no output init are needed -> deterministic.
//
// This revision replaces all int64 index handling with 32-bit loads/math:
// rows < 16384 and cols < 1e6, so the low dword of each little-endian int64 is
// the value, and the gather byte offset (col*256 + d*4 < 2^28) fits the GVS
// 32-bit VGPR offset -> single-VALU address math + b32 index loads.

typedef float v2f __attribute__((ext_vector_type(2)));
typedef float v8f __attribute__((ext_vector_type(8)));

#define EMBED_DIM 64
#define UNROLL 8   // 8 groups * 4 nnz = 32 nnz per main-loop iteration

__global__ void __launch_bounds__(128)
spmm_embed_wmma_kernel(const float* __restrict__ vals,
                       const float* __restrict__ emb,
                       const unsigned* __restrict__ rows32,  // low dwords of int64
                       const unsigned* __restrict__ cols32,  // low dwords of int64
                       float* __restrict__ out,
                       int nnz)
{
    const int tile  = blockIdx.x;          // 16 output rows: [tile*16, tile*16+16)
    const int wave  = threadIdx.x >> 5;    // 0..3 -> dim slice base = wave*16
    const int lane  = threadIdx.x & 31;
    const int mn    = lane & 15;           // M index for A, N index for B/C/D
    const int half  = lane >> 4;           // selects K pair {0,1} vs {2,3}

    const unsigned rlo   = (unsigned)tile * 16u;
    const unsigned rhi   = rlo + 16u;
    const unsigned myrow = rlo + (unsigned)mn;

    // rows[] is sorted: binary-search the nnz span owned by this row tile.
    // Uniform across the block; probes are broadcast loads.
    int lo = 0, hi = nnz;
    while (lo < hi) { int mid = (lo + hi) >> 1; if (rows32[2u * (unsigned)mid] < rlo) lo = mid + 1; else hi = mid; }
    const int start = lo;
    lo = start; hi = nnz;
    while (lo < hi) { int mid = (lo + hi) >> 1; if (rows32[2u * (unsigned)mid] < rhi) lo = mid + 1; else hi = mid; }
    const int end = lo;

    const unsigned dbase = (unsigned)(wave * 16 + mn); // this lane's embedding dim (N)

    v8f acc = {};                          // C/D: 16x16 f32 accumulator tile

    int i = start;

    // ---- main loop: 32 nnz / iteration; batch all gathers before the WMMAs
    // to keep ~2KB per wave in flight (4096 waves total -> ~8MB aggregate,
    // covering the HBM bandwidth-delay product).
    for (; i + (UNROLL * 4) <= end; i += UNROLL * 4) {
        v2f a[UNROLL], b[UNROLL];
#pragma unroll
        for (int g = 0; g < UNROLL; ++g) {
            const unsigned k0 = (unsigned)(i + 4 * g) + ((unsigned)half << 1);
            const unsigned r0 = rows32[2u * k0],     r1 = rows32[2u * k0 + 2u];
            const unsigned c0 = cols32[2u * k0],     c1 = cols32[2u * k0 + 2u];
            const float    v0 = vals[k0],            v1 = vals[k0 + 1u];
            // A (16x4): VGPR0 = K {0|2}, VGPR1 = K {1|3}; M = lane&15
            a[g].x = (r0 == myrow) ? v0 : 0.0f;
            a[g].y = (r1 == myrow) ? v1 : 0.0f;
            // B (4x16): same K striping; N = lane&15. 32-bit offsets -> GVS
            // addressing, 64B coalesced per gathered E row.
            b[g].x = emb[(c0 << 6) + dbase];
            b[g].y = emb[(c1 << 6) + dbase];
        }
#pragma unroll
        for (int g = 0; g < UNROLL; ++g) {
            // D = A*B + C  (f32, 16x16x4); chained accumulate on acc.
            acc = __builtin_amdgcn_wmma_f32_16x16x4_f32(
                false, a[g], false, b[g], (short)0, acc, false, false);
        }
    }

    // ---- tail: predicated 4-nnz groups; padded K lanes contribute A=0.
    for (; i < end; i += 4) {
        const int k0 = i + (half << 1);
        const int k1 = k0 + 1;
        v2f a; a.x = 0.0f; a.y = 0.0f;
        v2f b; b.x = 0.0f; b.y = 0.0f;
        if (k0 < end) {
            a.x = (rows32[2u * (unsigned)k0] == myrow) ? vals[k0] : 0.0f;
            b.x = emb[(cols32[2u * (unsigned)k0] << 6) + dbase];
        }
        if (k1 < end) {
            a.y = (rows32[2u * (unsigned)k1] == myrow) ? vals[k1] : 0.0f;
            b.y = emb[(cols32[2u * (unsigned)k1] << 6) + dbase];
        }
        // EXEC fully re-converged here (uniform loop bounds) as WMMA requires.
        acc = __builtin_amdgcn_wmma_f32_16x16x4_f32(
            false, a, false, b, (short)0, acc, false, false);
    }

    // ---- store D tile. C/D layout: VGPR v, lanes 0-15 -> M=v, lanes 16-31 ->
    // M=8+v, N=lane&15. Tiles are disjoint: plain stores, no init, deterministic.
#pragma unroll
    for (int v = 0; v < 8; ++v) {
        const unsigned m = (unsigned)(half * 8 + v);
        out[(size_t)(rlo + m) * EMBED_DIM + dbase] = acc[v];
    }
}

extern "C" void kernel_launch(void* const* d_in, const int* in_sizes, int n_in,
                              void* d_out, int out_size, void* d_ws, size_t ws_size,
                              hipStream_t stream) {
    const float*    vals   = (const float*)d_in[0];
    const float*    emb    = (const float*)d_in[1];
    const unsigned* rows32 = (const unsigned*)d_in[2];  // jnp.int64, low dwords
    const unsigned* cols32 = (const unsigned*)d_in[3];  // jnp.int64, low dwords
    float*          out    = (float*)d_out;

    const int nnz    = in_sizes[0];
    const int ntiles = out_size / (EMBED_DIM * 16);     // BATCH/16 = 1024

    spmm_embed_wmma_kernel<<<dim3(ntiles), dim3(128), 0, stream>>>(
        vals, emb, rows32, cols32, out, nnz);
}